// SpectralConvS2_72791105733243
// MI455X (gfx1250) — compile-verified
//
#include <hip/hip_runtime.h>

// Spherical spectral conv (SpectralConvS2) for gfx1250 / MI455X.
// All five stages expressed as batched GEMMs on V_WMMA_F32_16X16X4_F32,
// with cooperative, double-buffered LDS staging (global prefetch-to-regs
// overlapped with WMMA; ds_load_b64 fragment reads, bank-conflict free via
// 32-bank padded K-pair stride). Intermediates (2 x 72MB) ping-pong in d_ws
// and stay resident in the 192MB global L2.

typedef __attribute__((ext_vector_type(2))) float v2f;
typedef __attribute__((ext_vector_type(8))) float v8f;

#define NLATC 181
#define NLONC 360
#define LMAXC 181
#define MP    192                  // padded 181 -> 192 (12 tiles of 16)
#define NPAD  384                  // padded 360 -> 384 (6 col-blocks of 64)
#define BCC   256                  // B * CIN = 2 * 128
#define SS_   49152LL              // plane (re/im) stride   = 256*192
#define SM_   98304LL              // mode slab stride       = 2*SS_
#define BUFSZ 18874368LL           // floats per big buffer  = 192*SM_
#define FWD_ELEMS (2LL*192*360)
#define INV_ELEMS (2LL*192*384)
#define PI_D 3.14159265358979323846

// LDS tile geometry: block tile 128(rows) x 64(cols), K-chunk = 16.
// A stored as [kpair][row][2] with 288-word stride per kpair (256 + 32 pad),
// B stored as [kpair][col][2] with 160-word stride per kpair (128 + 32 pad).
// 288 % 64 == 32 and 160 % 64 == 32, so the two half-waves (which read
// adjacent K-pairs per the WMMA A/B VGPR layout) hit complementary bank sets.
// Double-buffered: per-buffer word count (ldsW) is a multiple of 64, so the
// bank mapping is identical in both buffers.
#define APW 288
#define BPW 160
#define A_WORDS (8*APW)            // 2304
#define B_WORDS (8*BPW)            // 1280
#define BUF1_WORDS (A_WORDS + B_WORDS)      // 3584 (pair1 only)
#define BUF2_WORDS (2*(A_WORDS + B_WORDS))  // 7168 (pair1 + pair2)

struct GemmDesc {
  const float* A; const float* B; const float* A2; const float* B2;
  float* D;
  long long sAb1, sAb2, sAr, sAk;       // A: batch-split strides, row, k
  long long sBb1, sBb2, sBk, sBc;       // B: batch-split strides, k, col
  long long sDb1, sDb2, sDr1, sDr2, sDc;// D: batch-split, row-split, col
  int bdiv;                             // batch t -> (t/bdiv, t%bdiv)
  int rdiv, rlim;                       // store row r -> (r/rdiv, r%rdiv); valid if (r%rdiv)<rlim
  int Rows, Cols, Kdim, ColsB, Cstore;
  int hasPair2; float s2;               // D = A*B + s2 * A2*B2
  int ldsW;                             // words per LDS double-buffer half
};

// Generic batched GEMM: block = 8 waves, block tile 128x64,
// wave tile 32x32 = 2x2 WMMA 16x16 tiles, double-buffered LDS staging.
__global__ __launch_bounds__(256) void sconv_wmma_gemm(GemmDesc g) {
  extern __shared__ float smem[];

  const int t  = blockIdx.z;
  const int tq = t / g.bdiv, tr = t % g.bdiv;
  const long long offA = (long long)tq * g.sAb1 + (long long)tr * g.sAb2;
  const long long offB = (long long)tq * g.sBb1 + (long long)tr * g.sBb2;
  const long long offD = (long long)tq * g.sDb1 + (long long)tr * g.sDb2;

  const int tid  = threadIdx.x;
  const int lane = tid & 31;
  const int wv   = tid >> 5;
  const int rblk = blockIdx.x * 128;
  const int cblk = blockIdx.y * 64;
  const int lrow = lane & 15;
  const int half = lane >> 4;              // 0: K {0,1}; 1: K {2,3}

  // Staging thread coordinates: kth fastest (coalesced for unit sBk/sAk).
  const int kth = tid & 15;
  const int rth = tid >> 4;                // 0..15
  const long long aG = offA + (long long)(rblk + rth) * g.sAr + (long long)kth * g.sAk;
  const long long bG = offB + (long long)kth * g.sBk + (long long)(cblk + rth) * g.sBc;
  const long long aStep  = 16 * g.sAr;     // per staging iteration (rows)
  const long long bStep  = 16 * g.sBc;     // per staging iteration (cols)
  const long long aKstep = 16 * g.sAk;     // per K-chunk
  const long long bKstep = 16 * g.sBk;
  const int aL = (kth >> 1) * APW + rth * 2 + (kth & 1);  // + it*32
  const int bL = (kth >> 1) * BPW + rth * 2 + (kth & 1);  // + it*32

  // Fragment LDS bases (word offsets), + ti*32 / tj*32, + p*APW / p*BPW.
  const int aF0 = ((wv & 3) * 32 + lrow) * 2;
  const int bF0 = ((wv >> 2) * 32 + lrow) * 2;
  const int row0 = rblk + (wv & 3) * 32;
  const int col0 = cblk + (wv >> 2) * 32;

  bool cok[4];
#pragma unroll
  for (int it = 0; it < 4; ++it)
    cok[it] = (cblk + rth + it * 16) < g.ColsB;

  float aV[8], bV[4], a2V[8], b2V[4];

  auto loadRegs = [&](int k0, long long aBase, long long bBase) {
    const bool kok = (k0 + kth) < g.Kdim;
#pragma unroll
    for (int it = 0; it < 8; ++it)
      aV[it] = kok ? g.A[aBase + it * aStep] : 0.0f;
#pragma unroll
    for (int it = 0; it < 4; ++it)
      bV[it] = (kok && cok[it]) ? g.B[bBase + it * bStep] : 0.0f;
    if (g.hasPair2) {
#pragma unroll
      for (int it = 0; it < 8; ++it)
        a2V[it] = kok ? g.s2 * g.A2[aBase + it * aStep] : 0.0f;
#pragma unroll
      for (int it = 0; it < 4; ++it)
        b2V[it] = (kok && cok[it]) ? g.B2[bBase + it * bStep] : 0.0f;
    }
  };

  auto storeRegs = [&](float* buf) {
    float* As_ = buf;
    float* Bs_ = buf + A_WORDS;
#pragma unroll
    for (int it = 0; it < 8; ++it) As_[aL + it * 32] = aV[it];
#pragma unroll
    for (int it = 0; it < 4; ++it) Bs_[bL + it * 32] = bV[it];
    if (g.hasPair2) {
      float* As2_ = buf + A_WORDS + B_WORDS;
      float* Bs2_ = As2_ + A_WORDS;
#pragma unroll
      for (int it = 0; it < 8; ++it) As2_[aL + it * 32] = a2V[it];
#pragma unroll
      for (int it = 0; it < 4; ++it) Bs2_[bL + it * 32] = b2V[it];
    }
  };

  v8f acc[2][2] = {};
  const int nchunks = (g.Kdim + 15) >> 4;

  // Prologue: stage chunk 0 into buffer 0.
  loadRegs(0, aG, bG);
  storeRegs(smem);
  __syncthreads();

  int curOff = 0;
  long long aP = aG, bP = bG;
  for (int i = 0; i < nchunks; ++i) {
    const bool hasNext = (i + 1) < nchunks;
    if (hasNext) {                        // prefetch next chunk into registers
      aP += aKstep; bP += bKstep;
      loadRegs((i + 1) * 16, aP, bP);
    }

    // Compute current chunk from LDS buffer `curOff`.
    float* Asc  = smem + curOff;
    float* Bsc  = Asc + A_WORDS;
    float* As2c = Bsc + B_WORDS;
    float* Bs2c = As2c + A_WORDS;
#pragma unroll
    for (int q = 0; q < 4; ++q) {
      const int p = q * 2 + half;          // K-pair index for this half-wave
      v2f a[2], b[2];
#pragma unroll
      for (int ti = 0; ti < 2; ++ti)
        a[ti] = *(const v2f*)&Asc[p * APW + aF0 + ti * 32];
#pragma unroll
      for (int tj = 0; tj < 2; ++tj)
        b[tj] = *(const v2f*)&Bsc[p * BPW + bF0 + tj * 32];
#pragma unroll
      for (int ti = 0; ti < 2; ++ti)
#pragma unroll
        for (int tj = 0; tj < 2; ++tj)
          acc[ti][tj] = __builtin_amdgcn_wmma_f32_16x16x4_f32(
              false, a[ti], false, b[tj], (short)0, acc[ti][tj], false, false);
      if (g.hasPair2) {
        v2f a2[2], b2[2];
#pragma unroll
        for (int ti = 0; ti < 2; ++ti)
          a2[ti] = *(const v2f*)&As2c[p * APW + aF0 + ti * 32];
#pragma unroll
        for (int tj = 0; tj < 2; ++tj)
          b2[tj] = *(const v2f*)&Bs2c[p * BPW + bF0 + tj * 32];
#pragma unroll
        for (int ti = 0; ti < 2; ++ti)
#pragma unroll
          for (int tj = 0; tj < 2; ++tj)
            acc[ti][tj] = __builtin_amdgcn_wmma_f32_16x16x4_f32(
                false, a2[ti], false, b2[tj], (short)0, acc[ti][tj], false, false);
      }
    }

    if (hasNext) storeRegs(smem + (g.ldsW - curOff));  // fill other buffer
    curOff = g.ldsW - curOff;
    __syncthreads();
  }

  // Store. C/D layout: lane<16 holds rows +{0..7}, lane>=16 holds +{8..15}.
#pragma unroll
  for (int ti = 0; ti < 2; ++ti)
#pragma unroll
    for (int tj = 0; tj < 2; ++tj) {
      const int c = col0 + tj * 16 + lrow;
      if (c < g.Cstore) {
        const int rbase = row0 + ti * 16 + half * 8;
        const long long cOff = offD + (long long)c * g.sDc;
#pragma unroll
        for (int e = 0; e < 8; ++e) {
          const int r  = rbase + e;
          const int rq = r / g.rdiv, rr = r % g.rdiv;
          if (rr < g.rlim)
            g.D[cOff + (long long)rq * g.sDr1 + (long long)rr * g.sDr2] = acc[ti][tj][e];
        }
      }
    }
}

// Forward DFT table  F[s][m][n], s=0: (2pi/360)cos, s=1: -(2pi/360)sin; m>=181 -> 0
__global__ void sconv_gen_fwd(float* F) {
  long long i = (long long)blockIdx.x * 256 + threadIdx.x;
  if (i >= FWD_ELEMS) return;
  int n = (int)(i % 360);
  int m = (int)((i / 360) % 192);
  int s = (int)(i / (360 * 192));
  float v = 0.0f;
  if (m < LMAXC) {
    double th = 2.0 * PI_D * (double)((m * n) % 360) / 360.0;
    double sc = 2.0 * PI_D / 360.0;
    v = (float)(s == 0 ? sc * cos(th) : -sc * sin(th));
  }
  F[i] = v;
}

// Inverse DFT table G[s][m][n], with hermitian weight a_m (1,2,...,2,1);
// s=0: a_m cos, s=1: -a_m sin; zero padding for m>=181 or n>=360.
__global__ void sconv_gen_inv(float* G) {
  long long i = (long long)blockIdx.x * 256 + threadIdx.x;
  if (i >= INV_ELEMS) return;
  int n = (int)(i % NPAD);
  int m = (int)((i / NPAD) % 192);
  int s = (int)(i / (NPAD * 192));
  float v = 0.0f;
  if (m < LMAXC && n < NLONC) {
    double a  = (m == 0 || m == 180) ? 1.0 : 2.0;
    double th = 2.0 * PI_D * (double)((m * n) % 360) / 360.0;
    v = (float)(s == 0 ? a * cos(th) : -a * sin(th));
  }
  G[i] = v;
}

extern "C" void kernel_launch(void* const* d_in, const int* in_sizes, int n_in,
                              void* d_out, int out_size, void* d_ws, size_t ws_size,
                              hipStream_t stream) {
  (void)in_sizes; (void)n_in; (void)out_size; (void)ws_size;
  const float* x   = (const float*)d_in[0];   // (2,128,181,360)
  const float* wr  = (const float*)d_in[1];   // (1,128,128,181)
  const float* wi  = (const float*)d_in[2];   // (1,128,128,181)
  const float* sht = (const float*)d_in[3];   // (181,181,181) [m][l][k]
  const float* pct = (const float*)d_in[4];   // (181,181,181) [m][l][k]
  float* out = (float*)d_out;

  float* ws   = (float*)d_ws;
  float* bufA = ws;                       // XF, later Y     [m][s][bc][*] padded
  float* bufB = ws + BUFSZ;               // coeffs, later YK
  float* FWD  = ws + 2 * BUFSZ;           // [2][192][360]
  float* INV  = FWD + FWD_ELEMS;          // [2][192][384]
  // total ws use: (2*BUFSZ + FWD + INV)*4 ~= 152.1 MB

  const size_t smem1 = (size_t)(2 * BUF1_WORDS) * 4;  // 28672 B (double-buffered)
  const size_t smem2 = (size_t)(2 * BUF2_WORDS) * 4;  // 57344 B (double-buffered)

  // Zero the two big buffers so all padded slabs (m>=181, k>=181) are 0.
  hipMemsetAsync(d_ws, 0, (size_t)(2 * BUFSZ) * sizeof(float), stream);
  sconv_gen_fwd<<<(int)((FWD_ELEMS + 255) / 256), 256, 0, stream>>>(FWD);
  sconv_gen_inv<<<(int)((INV_ELEMS + 255) / 256), 256, 0, stream>>>(INV);

  // ---- Stage 1: XF[m][s][bc][k] = sum_n x[bc,k,n] * F[s][m][n] ----
  {
    GemmDesc g = {};
    g.A = x; g.B = FWD; g.D = bufA;
    g.sAb1 = 0; g.sAb2 = 0; g.sAr = 360; g.sAk = 1;
    g.sBb1 = 0; g.sBb2 = 192LL * 360; g.sBk = 1; g.sBc = 360;
    g.sDb1 = 0; g.sDb2 = SS_; g.sDr1 = 192; g.sDr2 = 1; g.sDc = SM_;
    g.bdiv = 2; g.rdiv = 181; g.rlim = 181;
    g.Rows = 46336; g.Cols = MP; g.Kdim = 360; g.ColsB = MP; g.Cstore = MP;
    g.hasPair2 = 0; g.s2 = 0.0f; g.ldsW = BUF1_WORDS;
    sconv_wmma_gemm<<<dim3(362, 3, 2), 256, smem1, stream>>>(g);
  }

  // ---- Stage 2: C2[m][s][bc][l] = sum_k XF[m][s][bc][k] * sht[m][l][k] ----
  {
    GemmDesc g = {};
    g.A = bufA; g.B = sht; g.D = bufB;
    g.sAb1 = SM_; g.sAb2 = SS_; g.sAr = 192; g.sAk = 1;
    g.sBb1 = 181LL * 181; g.sBb2 = 0; g.sBk = 1; g.sBc = 181;
    g.sDb1 = SM_; g.sDb2 = SS_; g.sDr1 = 0; g.sDr2 = 192; g.sDc = 1;
    g.bdiv = 2; g.rdiv = 1 << 30; g.rlim = 1 << 30;
    g.Rows = BCC; g.Cols = MP; g.Kdim = LMAXC; g.ColsB = LMAXC; g.Cstore = MP;
    g.hasPair2 = 0; g.s2 = 0.0f; g.ldsW = BUF1_WORDS;
    sconv_wmma_gemm<<<dim3(2, 3, 362), 256, smem1, stream>>>(g);
  }

  // ---- Stage 3: complex channel mix, batched over t=(b,l), b=t/181, l=t%181
  //   Yre[m][0][b*128+o][l] = sum_i wr[i,o,l]*C2re - wi[i,o,l]*C2im
  //   Yim[m][1][b*128+o][l] = sum_i wi[i,o,l]*C2re + wr[i,o,l]*C2im
  {
    GemmDesc g = {};
    g.sAb1 = 0; g.sAb2 = 1; g.sAr = 181; g.sAk = 128LL * 181;
    g.sBb1 = 128LL * 192; g.sBb2 = 1; g.sBk = 192; g.sBc = SM_;
    g.sDb1 = 128LL * 192; g.sDb2 = 1; g.sDr1 = 0; g.sDr2 = 192; g.sDc = SM_;
    g.bdiv = 181; g.rdiv = 1 << 30; g.rlim = 1 << 30;
    g.Rows = 128; g.Cols = MP; g.Kdim = 128; g.ColsB = MP; g.Cstore = MP;
    g.hasPair2 = 1; g.ldsW = BUF2_WORDS;

    GemmDesc gre = g;
    gre.A = wr; gre.B = bufB; gre.A2 = wi; gre.B2 = bufB + SS_;
    gre.D = bufA; gre.s2 = -1.0f;
    sconv_wmma_gemm<<<dim3(1, 3, 362), 256, smem2, stream>>>(gre);

    GemmDesc gim = g;
    gim.A = wi; gim.B = bufB; gim.A2 = wr; gim.B2 = bufB + SS_;
    gim.D = bufA + SS_; gim.s2 = 1.0f;
    sconv_wmma_gemm<<<dim3(1, 3, 362), 256, smem2, stream>>>(gim);
  }

  // ---- Stage 4: YK[m][s][bc][k] = sum_l Y[m][s][bc][l] * pct[m][l][k] ----
  {
    GemmDesc g = {};
    g.A = bufA; g.B = pct; g.D = bufB;
    g.sAb1 = SM_; g.sAb2 = SS_; g.sAr = 192; g.sAk = 1;
    g.sBb1 = 181LL * 181; g.sBb2 = 0; g.sBk = 181; g.sBc = 1;
    g.sDb1 = SM_; g.sDb2 = SS_; g.sDr1 = 0; g.sDr2 = 192; g.sDc = 1;
    g.bdiv = 2; g.rdiv = 1 << 30; g.rlim = 1 << 30;
    g.Rows = BCC; g.Cols = MP; g.Kdim = LMAXC; g.ColsB = LMAXC; g.Cstore = MP;
    g.hasPair2 = 0; g.s2 = 0.0f; g.ldsW = BUF1_WORDS;
    sconv_wmma_gemm<<<dim3(2, 3, 362), 256, smem1, stream>>>(g);
  }

  // ---- Stage 5: out[bc,k,n] = sum_m YKre*Gc[m][n] + YKim*Gs[m][n]
  //   rows r = bc*192 + k (A is contiguous in r); store skips k>=181.
  {
    GemmDesc g = {};
    g.A = bufB; g.B = INV; g.A2 = bufB + SS_; g.B2 = INV + 192LL * NPAD;
    g.D = out;
    g.sAb1 = 0; g.sAb2 = 0; g.sAr = 1; g.sAk = SM_;
    g.sBb1 = 0; g.sBb2 = 0; g.sBk = NPAD; g.sBc = 1;
    g.sDb1 = 0; g.sDb2 = 0; g.sDr1 = 181LL * 360; g.sDr2 = 360; g.sDc = 1;
    g.bdiv = 1; g.rdiv = 192; g.rlim = 181;
    g.Rows = 49152; g.Cols = NPAD; g.Kdim = MP; g.ColsB = NPAD; g.Cstore = NLONC;
    g.hasPair2 = 1; g.s2 = 1.0f; g.ldsW = BUF2_WORDS;
    sconv_wmma_gemm<<<dim3(384, 6, 1), 256, smem2, stream>>>(g);
  }
}